// SuperQ_76106820485392
// MI455X (gfx1250) — compile-verified
//
#include <hip/hip_runtime.h>
#include <hip/hip_bf16.h>
#include <stdint.h>

// Superquadric truncated-SDF for 2M points / 32 SQs.
// Bandwidth-bound: 256MB one-hot + 24MB points + 8MB out ~ 12.4us @ 23.3TB/s.
// CDNA5 paths: async DMA of the SQ parameter table into LDS
// (global_load_async_to_lds_b32 / s_wait_asynccnt), ds_load param gathers,
// non-temporal (th:NT) streaming for the single-touch bulk data.

#define MAXSQ  32
#define PSTR   21              // 21 dwords/SQ record: coprime w/ 64 banks
#define TRUNC  0.1f

typedef int   v4i __attribute__((ext_vector_type(4)));
typedef float v4f __attribute__((ext_vector_type(4)));

// record layout (floats):
// [0..2]=b1 [3..5]=b2 [6..8]=b3 [9..11]=t [12..14]=1/s
// [15]=1/e2 [16]=1/e1 [17]=e2/e1 [18]=-e1/2 [19..20]=pad

__device__ __forceinline__ void sq_make_params(
    int s, const float* __restrict__ raw_sqscale, const float* __restrict__ raw_exponents,
    const float* __restrict__ raw_rotation, const float* __restrict__ translation,
    float* __restrict__ o)
{
    // raw_rotation layout (nsq,3,2): a1 = col0, a2 = col1
    float a1x = raw_rotation[s*6+0], a2x = raw_rotation[s*6+1];
    float a1y = raw_rotation[s*6+2], a2y = raw_rotation[s*6+3];
    float a1z = raw_rotation[s*6+4], a2z = raw_rotation[s*6+5];
    float n1 = fmaxf(sqrtf(a1x*a1x + a1y*a1y + a1z*a1z), 1e-12f);
    float b1x = a1x/n1, b1y = a1y/n1, b1z = a1z/n1;
    float dot = b1x*a2x + b1y*a2y + b1z*a2z;
    float c2x = a2x - dot*b1x, c2y = a2y - dot*b1y, c2z = a2z - dot*b1z;
    float n2 = fmaxf(sqrtf(c2x*c2x + c2y*c2y + c2z*c2z), 1e-12f);
    float b2x = c2x/n2, b2y = c2y/n2, b2z = c2z/n2;
    float b3x = b1y*b2z - b1z*b2y;
    float b3y = b1z*b2x - b1x*b2z;
    float b3z = b1x*b2y - b1y*b2x;
    float s0 = fmaxf(raw_sqscale[s*3+0], 1e-5f);
    float s1 = fmaxf(raw_sqscale[s*3+1], 1e-5f);
    float s2 = fmaxf(raw_sqscale[s*3+2], 1e-5f);
    float e1 = fmaxf(raw_exponents[s*2+0], 0.01f);
    float e2 = fmaxf(raw_exponents[s*2+1], 0.01f);
    o[0]=b1x; o[1]=b1y; o[2]=b1z;
    o[3]=b2x; o[4]=b2y; o[5]=b2z;
    o[6]=b3x; o[7]=b3y; o[8]=b3z;
    o[9]=translation[s*3+0]; o[10]=translation[s*3+1]; o[11]=translation[s*3+2];
    o[12]=1.0f/s0; o[13]=1.0f/s1; o[14]=1.0f/s2;
    o[15]=1.0f/e2; o[16]=1.0f/e1; o[17]=e2/e1; o[18]=-0.5f*e1;
    o[19]=0.0f; o[20]=0.0f;
}

__global__ void sq_prep_kernel(
    const float* __restrict__ raw_sqscale, const float* __restrict__ raw_exponents,
    const float* __restrict__ raw_rotation, const float* __restrict__ translation,
    float* __restrict__ wsParams, int nsq)
{
    int s = threadIdx.x;
    if (s < nsq) {
        sq_make_params(s, raw_sqscale, raw_exponents, raw_rotation, translation,
                       wsParams + (size_t)s * PSTR);
    }
}

// Stage the parameter table into LDS. Async DMA path when wsParams != null.
__device__ __forceinline__ void sq_stage_lds(
    float* __restrict__ P, const float* __restrict__ wsParams,
    const float* __restrict__ raw_sqscale, const float* __restrict__ raw_exponents,
    const float* __restrict__ raw_rotation, const float* __restrict__ translation,
    int nsq)
{
    if (wsParams) {
        int total = nsq * PSTR;
        for (int idx = threadIdx.x; idx < total; idx += blockDim.x) {
            unsigned ldsOff = (unsigned)(uintptr_t)(&P[idx]);   // low 32b of generic ptr = LDS offset
            unsigned gOff   = (unsigned)(idx * 4);
            asm volatile("global_load_async_to_lds_b32 %0, %1, %2"
                         :: "v"(ldsOff), "v"(gOff), "s"(wsParams)
                         : "memory");
        }
        asm volatile("s_wait_asynccnt 0x0" ::: "memory");
    } else {
        if (threadIdx.x < nsq) {
            sq_make_params((int)threadIdx.x, raw_sqscale, raw_exponents, raw_rotation,
                           translation, P + threadIdx.x * PSTR);
        }
    }
    __syncthreads();
}

__device__ __forceinline__ float sq_eval(const float* __restrict__ P, int seg, int assigned,
                                         float px, float py, float pz)
{
    const float* o = P + seg * PSTR;
    float cx = px - o[9], cy = py - o[10], cz = pz - o[11];
    float X0 = o[0]*cx + o[1]*cy + o[2]*cz;   // b1 . pc
    float X1 = o[3]*cx + o[4]*cy + o[5]*cz;   // b2 . pc
    float X2 = o[6]*cx + o[7]*cy + o[8]*cz;   // b3 . pc
    float a0 = fmaxf(fabsf(X0), 1e-6f); X0 = (X0 > 0.0f) ? a0 : -a0;
    float a1 = fmaxf(fabsf(X1), 1e-6f); X1 = (X1 > 0.0f) ? a1 : -a1;
    float a2 = fmaxf(fabsf(X2), 1e-6f); X2 = (X2 > 0.0f) ? a2 : -a2;
    float r0 = sqrtf(X0*X0 + X1*X1 + X2*X2);
    float u0 = X0 * o[12]; u0 *= u0;
    float u1 = X1 * o[13]; u1 *= u1;
    float u2 = X2 * o[14]; u2 *= u2;
    float ie2 = o[15], ie1 = o[16], e21 = o[17], nh = o[18];
    float t1 = powf(u0, ie2);
    float t2 = powf(u1, ie2);
    float t3 = powf(u2, ie1);
    float f  = powf(powf(t1 + t2, e21) + t3, nh);
    float sdf = r0 * (1.0f - f);
    sdf = fminf(fmaxf(sdf, -TRUNC), TRUNC);
    return assigned ? sdf : 0.0f;
}

// Vectorized path: 4 points/thread, b128 non-temporal streams. Requires n % 4 == 0.
__global__ __launch_bounds__(256) void sq_sdf_vec4_kernel(
    const float* __restrict__ points, const int* __restrict__ assign,
    const float* __restrict__ wsParams,
    const float* __restrict__ raw_sqscale, const float* __restrict__ raw_exponents,
    const float* __restrict__ raw_rotation, const float* __restrict__ translation,
    float* __restrict__ out, int n, int nsq)
{
    __shared__ float P[MAXSQ * PSTR];
    sq_stage_lds(P, wsParams, raw_sqscale, raw_exponents, raw_rotation, translation, nsq);

    int t  = blockIdx.x * blockDim.x + threadIdx.x;
    int p0 = t * 4;
    if (p0 >= n) return;

    // argmax over one-hot rows (first-max semantics), per 4 points.
    // unroll 8 -> clause 8 b128 loads in flight per trip (MLP for the 256MB stream)
    int bx = 0, by = 0, bz = 0, bw = 0;
    int sx = 0, sy = 0, sz = 0, sw = 0;
    const v4i* amv = (const v4i*)assign;
    size_t base = (size_t)p0 >> 2;
    size_t rowstride = (size_t)n >> 2;
    #pragma unroll 8
    for (int r = 0; r < nsq; ++r) {
        v4i v = __builtin_nontemporal_load(&amv[base + (size_t)r * rowstride]);
        if (v.x > bx) { bx = v.x; sx = r; }
        if (v.y > by) { by = v.y; sy = r; }
        if (v.z > bz) { bz = v.z; sz = r; }
        if (v.w > bw) { bw = v.w; sw = r; }
    }

    // 12 packed coords for 4 points: three b128 NT loads (16B aligned: 48*t bytes)
    const v4f* pv = (const v4f*)points;
    size_t pb = (size_t)t * 3;
    v4f q0 = __builtin_nontemporal_load(&pv[pb + 0]);
    v4f q1 = __builtin_nontemporal_load(&pv[pb + 1]);
    v4f q2 = __builtin_nontemporal_load(&pv[pb + 2]);

    v4f res;
    res.x = sq_eval(P, sx, bx == 1, q0.x, q0.y, q0.z);
    res.y = sq_eval(P, sy, by == 1, q0.w, q1.x, q1.y);
    res.z = sq_eval(P, sz, bz == 1, q1.z, q1.w, q2.x);
    res.w = sq_eval(P, sw, bw == 1, q2.y, q2.z, q2.w);

    __builtin_nontemporal_store(res, &((v4f*)out)[t]);
}

// Scalar fallback (any n).
__global__ __launch_bounds__(256) void sq_sdf_scalar_kernel(
    const float* __restrict__ points, const int* __restrict__ assign,
    const float* __restrict__ wsParams,
    const float* __restrict__ raw_sqscale, const float* __restrict__ raw_exponents,
    const float* __restrict__ raw_rotation, const float* __restrict__ translation,
    float* __restrict__ out, int n, int nsq)
{
    __shared__ float P[MAXSQ * PSTR];
    sq_stage_lds(P, wsParams, raw_sqscale, raw_exponents, raw_rotation, translation, nsq);

    int p = blockIdx.x * blockDim.x + threadIdx.x;
    if (p >= n) return;

    int best = 0, seg = 0;
    #pragma unroll 8
    for (int r = 0; r < nsq; ++r) {
        int v = __builtin_nontemporal_load(&assign[(size_t)r * (size_t)n + (size_t)p]);
        if (v > best) { best = v; seg = r; }
    }
    float px = __builtin_nontemporal_load(&points[(size_t)p * 3 + 0]);
    float py = __builtin_nontemporal_load(&points[(size_t)p * 3 + 1]);
    float pz = __builtin_nontemporal_load(&points[(size_t)p * 3 + 2]);
    float r  = sq_eval(P, seg, best == 1, px, py, pz);
    __builtin_nontemporal_store(r, &out[p]);
}

extern "C" void kernel_launch(void* const* d_in, const int* in_sizes, int n_in,
                              void* d_out, int out_size, void* d_ws, size_t ws_size,
                              hipStream_t stream)
{
    const float* points        = (const float*)d_in[0];
    const float* raw_sqscale   = (const float*)d_in[1];
    const float* raw_exponents = (const float*)d_in[2];
    const float* raw_rotation  = (const float*)d_in[3];
    const float* translation   = (const float*)d_in[4];
    const int*   assign        = (const int*)d_in[5];
    float*       out           = (float*)d_out;

    int n   = in_sizes[0] / 3;   // number of points
    int nsq = in_sizes[1] / 3;   // number of superquadrics (32)
    if (nsq > MAXSQ) nsq = MAXSQ;

    float* wsParams = nullptr;
    size_t need = (size_t)nsq * PSTR * sizeof(float);
    if (d_ws && ws_size >= need) wsParams = (float*)d_ws;

    if (wsParams) {
        sq_prep_kernel<<<1, MAXSQ, 0, stream>>>(raw_sqscale, raw_exponents,
                                                raw_rotation, translation, wsParams, nsq);
    }

    if ((n & 3) == 0) {
        int nthreads = n / 4;
        int blocks = (nthreads + 255) / 256;
        sq_sdf_vec4_kernel<<<blocks, 256, 0, stream>>>(
            points, assign, wsParams, raw_sqscale, raw_exponents, raw_rotation,
            translation, out, n, nsq);
    } else {
        int blocks = (n + 255) / 256;
        sq_sdf_scalar_kernel<<<blocks, 256, 0, stream>>>(
            points, assign, wsParams, raw_sqscale, raw_exponents, raw_rotation,
            translation, out, n, nsq);
    }
}